// RBLNTopKTopPSampler_26104811225233
// MI455X (gfx1250) — compile-verified
//
#include <hip/hip_runtime.h>
#include <hip/hip_bf16.h>
#include <math.h>

// ---------------------------------------------------------------------------
// Exact top-k/top-p sampler for B=256, V=128000, k<=64.
//   Pass 1 (TDM-streamed): row max + 4096-bin radix histogram (monotone key).
//   Pass 2 (TDM-streamed): softmax denominator + gather candidates >= rank-64
//                          threshold bin (~100-200 elements for N(0,2) logits).
//   Then: in-LDS bitonic sort (stable: value desc, index asc), top-k/top-p
//   masks on the top-64, renormalize, JAX threefry2x32 Gumbel-max sample.
// Data movement: 2 x 131MB (2nd pass mostly L2-resident) ~= 9us at 23.3TB/s.
// CDNA5 paths: tensor_load_to_lds (TDM double buffering, confirmed by vmem=4
// in the histogram: logits never touch per-lane global loads), s_wait_tensorcnt,
// wave32 shuffle reductions, ds_load_b128 tile consumption.
// ---------------------------------------------------------------------------

#define BROWS    256
#define VOCAB    128000
#define TILE     4096
#define NTILES   ((VOCAB + TILE - 1) / TILE)   // 32 (last tile = 1024)
#define NTHREADS 512
#define NWAVES   (NTHREADS / 32)
#define CAND_MAX 1024

typedef unsigned int v4u __attribute__((ext_vector_type(4)));
typedef int          v4i __attribute__((ext_vector_type(4)));
typedef int          v8i __attribute__((ext_vector_type(8)));
typedef float        f4  __attribute__((ext_vector_type(4)));

__device__ __forceinline__ int imin(int a, int b) { return a < b ? a : b; }

// Monotone float->uint key: ascending key == ascending float.
__device__ __forceinline__ unsigned mono_key(float x) {
  unsigned b = __float_as_uint(x);
  return (b & 0x80000000u) ? ~b : (b | 0x80000000u);
}
__device__ __forceinline__ float inv_mono(unsigned u) {
  unsigned b = (u & 0x80000000u) ? (u & 0x7FFFFFFFu) : ~u;
  return __uint_as_float(b);
}

// ---- Tensor Data Mover: 1-D row tile (n float32) global -> LDS ------------
__device__ __forceinline__ void tdm_load_tile(unsigned lds_byte_off,
                                              const float* gptr, unsigned n) {
  unsigned long long ga = (unsigned long long)(size_t)gptr;
  v4u g0;
  g0[0] = 1u;                                            // count=1, user D#
  g0[1] = lds_byte_off;                                  // lds_addr (bytes)
  g0[2] = (unsigned)(ga & 0xFFFFFFFFull);                // global_addr[31:0]
  g0[3] = (unsigned)((ga >> 32) & 0x01FFFFFFull)         // global_addr[56:32]
        | (2u << 30);                                    // type=2 ("image")
  v8i g1;
  g1[0] = (int)(2u << 16);                               // data_size=2 (4B)
  g1[1] = (int)((n & 0xFFFFu) << 16);                    // tensor_dim0[15:0]
  g1[2] = (int)(((n >> 16) & 0xFFFFu) | (1u << 16));     // dim0 hi | tensor_dim1=1
  g1[3] = (int)((n & 0xFFFFu) << 16);                    // tile_dim0 = n
  g1[4] = 1;                                             // tile_dim1=1, tile_dim2=0
  g1[5] = (int)n;                                        // tensor_dim0_stride
  g1[6] = 0;
  g1[7] = 0;
  v4i g2 = {0, 0, 0, 0};
  v4i g3 = {0, 0, 0, 0};
#if __has_include(<hip/amd_detail/amd_gfx1250_TDM.h>)
  // amdgpu-toolchain (clang-23 / therock headers): 6-arg builtin
  v8i g4 = {0, 0, 0, 0, 0, 0, 0, 0};
  __builtin_amdgcn_tensor_load_to_lds(g0, g1, g2, g3, g4, 0);
#else
  // ROCm 7.2 (clang-22): 5-arg builtin
  __builtin_amdgcn_tensor_load_to_lds(g0, g1, g2, g3, 0);
#endif
}

// ---- JAX threefry2x32 with key(42) = (0, 42) ------------------------------
__device__ __forceinline__ unsigned rotl32(unsigned x, int d) {
  return (x << d) | (x >> (32 - d));
}
__device__ unsigned threefry_bits(unsigned idx) {
  const unsigned half = (unsigned)BROWS * (unsigned)VOCAB / 2u;  // 16384000
  unsigned x0, x1;
  bool second;
  if (idx < half) { x0 = idx;        x1 = idx + half; second = false; }
  else            { x0 = idx - half; x1 = idx;        second = true;  }
  unsigned ks[3] = {0u, 42u, 0u ^ 42u ^ 0x1BD11BDAu};
  x0 += ks[0]; x1 += ks[1];
  const int rotA[4] = {13, 15, 26, 6};
  const int rotB[4] = {17, 29, 16, 24};
  for (int i = 0; i < 5; ++i) {
    const int* r = (i & 1) ? rotB : rotA;
#pragma unroll
    for (int j = 0; j < 4; ++j) { x0 += x1; x1 = rotl32(x1, r[j]); x1 ^= x0; }
    x0 += ks[(i + 1) % 3];
    x1 += ks[(i + 2) % 3] + (unsigned)(i + 1);
  }
  return second ? x1 : x0;
}
__device__ __forceinline__ float gumbel_at(unsigned idx) {
  unsigned bits = threefry_bits(idx);
  float f = __uint_as_float((bits >> 9) | 0x3F800000u) - 1.0f;  // [0,1)
  const float tiny = 1.17549435e-38f;
  float u = fmaxf(tiny, f * (1.0f - tiny) + tiny);
  return -logf(-logf(u));
}

__global__ void __launch_bounds__(NTHREADS, 1)
RBLNTopKTopPSampler_kernel(const float* __restrict__ logits,
                           const int* __restrict__ kArr,
                           const float* __restrict__ pArr,
                           float* __restrict__ out) {
  __shared__ __align__(16) float tileBuf[2][TILE];   // 32 KB, TDM double buffer
  __shared__ unsigned hist[4096];                    // 16 KB radix histogram
  __shared__ unsigned chunkSum[NTHREADS];            // 2 KB
  __shared__ unsigned long long cand[CAND_MAX];      // 8 KB (~key<<32 | idx)
  __shared__ float wred[NWAVES];
  __shared__ float sVal[64];
  __shared__ int   sTok[64];
  __shared__ float sProb[64];
  __shared__ float sScore[64];
  __shared__ int   sKeep[64];
  __shared__ float sM, sS, sZ;
  __shared__ unsigned sCandCount, sNc;
  __shared__ int sBthr;

  const int b   = blockIdx.x;
  const int tid = threadIdx.x;
  const int wave = tid >> 5;                 // wave32 on gfx1250
  const float* row = logits + (size_t)b * VOCAB;

  for (int i = tid; i < 4096; i += NTHREADS) hist[i] = 0u;
  if (tid == 0) sCandCount = 0u;
  __syncthreads();

  // ---------------- Pass 1: max + histogram (TDM streamed) ----------------
  float mloc = -INFINITY;
  if (wave == 0)
    tdm_load_tile((unsigned)(size_t)&tileBuf[0][0], row, (unsigned)imin(TILE, VOCAB));
  for (int t = 0; t < NTILES; ++t) {
    const int n = imin(TILE, VOCAB - t * TILE);
    if (wave == 0) __builtin_amdgcn_s_wait_tensorcnt(0);   // tile t landed
    __syncthreads();
    if (wave == 0 && t + 1 < NTILES) {                     // prefetch t+1
      const int n2 = imin(TILE, VOCAB - (t + 1) * TILE);
      tdm_load_tile((unsigned)(size_t)&tileBuf[(t + 1) & 1][0],
                    row + (size_t)(t + 1) * TILE, (unsigned)n2);
    }
    const float* tl = tileBuf[t & 1];
    for (int i = 4 * tid; i < n; i += 4 * NTHREADS) {      // ds_load_b128
      f4 x = *(const f4*)&tl[i];
      mloc = fmaxf(mloc, fmaxf(fmaxf(x.x, x.y), fmaxf(x.z, x.w)));
      atomicAdd(&hist[mono_key(x.x) >> 20], 1u);           // top 12 bits -> bin
      atomicAdd(&hist[mono_key(x.y) >> 20], 1u);
      atomicAdd(&hist[mono_key(x.z) >> 20], 1u);
      atomicAdd(&hist[mono_key(x.w) >> 20], 1u);
    }
    __syncthreads();                                       // done with buf[t&1]
  }
  {
    float wm = mloc;
    for (int off = 16; off > 0; off >>= 1) wm = fmaxf(wm, __shfl_xor(wm, off, 32));
    if ((tid & 31) == 0) wred[wave] = wm;
  }
  __syncthreads();
  if (tid == 0) {
    float m = -INFINITY;
    for (int i = 0; i < NWAVES; ++i) m = fmaxf(m, wred[i]);
    sM = m;
  }
  __syncthreads();
  const float M = sM;

  // ---------- threshold bin: smallest bin set covering rank 64 ------------
  {
    unsigned cs = 0u;
    for (int j = 0; j < 8; ++j) cs += hist[tid * 8 + j];
    chunkSum[tid] = cs;
  }
  __syncthreads();
  if (tid == 0) {
    unsigned acc = 0u;
    int c = NTHREADS - 1;
    for (; c >= 0; --c) { if (acc + chunkSum[c] >= 64u) break; acc += chunkSum[c]; }
    if (c < 0) c = 0;
    int bin = c * 8 + 7;
    for (; bin > c * 8; --bin) { if (acc + hist[bin] >= 64u) break; acc += hist[bin]; }
    sBthr = bin;
  }
  __syncthreads();
  const unsigned bthr = (unsigned)sBthr;

  // ------------- Pass 2: sumexp + candidate gather (TDM streamed) ---------
  float sloc = 0.0f;
  if (wave == 0)
    tdm_load_tile((unsigned)(size_t)&tileBuf[0][0], row, (unsigned)imin(TILE, VOCAB));
  for (int t = 0; t < NTILES; ++t) {
    const int n = imin(TILE, VOCAB - t * TILE);
    if (wave == 0) __builtin_amdgcn_s_wait_tensorcnt(0);
    __syncthreads();
    if (wave == 0 && t + 1 < NTILES) {
      const int n2 = imin(TILE, VOCAB - (t + 1) * TILE);
      tdm_load_tile((unsigned)(size_t)&tileBuf[(t + 1) & 1][0],
                    row + (size_t)(t + 1) * TILE, (unsigned)n2);
    }
    const float* tl = tileBuf[t & 1];
    for (int i = 4 * tid; i < n; i += 4 * NTHREADS) {      // ds_load_b128
      f4 x = *(const f4*)&tl[i];
      sloc += expf(x.x - M) + expf(x.y - M) + expf(x.z - M) + expf(x.w - M);
#pragma unroll
      for (int e = 0; e < 4; ++e) {
        float xv = (e == 0) ? x.x : (e == 1) ? x.y : (e == 2) ? x.z : x.w;
        unsigned key = mono_key(xv);
        if ((key >> 20) >= bthr) {
          unsigned pos = atomicAdd(&sCandCount, 1u);
          if (pos < CAND_MAX)
            cand[pos] = ((unsigned long long)(~key) << 32) |
                        (unsigned long long)(unsigned)(t * TILE + i + e);
        }
      }
    }
    __syncthreads();
  }
  {
    float ws = sloc;
    for (int off = 16; off > 0; off >>= 1) ws += __shfl_xor(ws, off, 32);
    if ((tid & 31) == 0) wred[wave] = ws;
  }
  __syncthreads();
  if (tid == 0) {
    float s2 = 0.0f;
    for (int i = 0; i < NWAVES; ++i) s2 += wred[i];
    sS = s2;
    sNc = sCandCount < CAND_MAX ? sCandCount : CAND_MAX;
  }
  __syncthreads();
  const float S = sS;
  const int nc = (int)sNc;

  // --------- bitonic sort candidates: value desc, index asc (stable) ------
  int P = 64;
  while (P < nc) P <<= 1;                               // <= CAND_MAX
  for (int i = nc + tid; i < P; i += NTHREADS) cand[i] = ~0ull;
  __syncthreads();
  for (int kk = 2; kk <= P; kk <<= 1) {
    for (int j = kk >> 1; j > 0; j >>= 1) {
      for (int i = tid; i < P; i += NTHREADS) {
        int partner = i ^ j;
        if (partner > i) {
          unsigned long long a = cand[i], c2 = cand[partner];
          bool up = ((i & kk) == 0);
          if ((a > c2) == up) { cand[i] = c2; cand[partner] = a; }
        }
      }
      __syncthreads();
    }
  }

  // ---------------- top-64: probs, masks, renormalize ---------------------
  if (tid < 64) {
    if (tid < nc) {
      unsigned long long pk = cand[tid];
      unsigned key = ~(unsigned)(pk >> 32);
      sVal[tid]  = inv_mono(key);
      sTok[tid]  = (int)(unsigned)(pk & 0xFFFFFFFFull);
      sProb[tid] = expf(sVal[tid] - M) / S;
    } else {
      sVal[tid] = -INFINITY; sTok[tid] = 0; sProb[tid] = 0.0f;
    }
  }
  __syncthreads();
  if (tid == 0) {
    int   krow = kArr[b];
    float prow = pArr[b];
    float c = 0.0f, Z = 0.0f;
    for (int r = 0; r < 64; ++r) {
      float spv = sProb[r];
      int kept = (r < krow) && (c < prow) && (r < nc);  // cum-mass BEFORE r < p
      sKeep[r] = kept;
      if (kept) Z += spv;
      c += spv;
    }
    sZ = Z;
  }
  __syncthreads();
  // ------------- Gumbel-max sample (threefry at flat b*V + r) -------------
  if (tid < 64) {
    if (sKeep[tid]) {
      float fp = sProb[tid] / sZ;
      float lp = logf(fmaxf(fp, 1e-38f));
      sScore[tid] = lp + gumbel_at((unsigned)(b * VOCAB + tid));
    } else {
      sScore[tid] = -INFINITY;
    }
  }
  __syncthreads();
  if (tid == 0) {
    float best = -INFINITY;
    int bi = 0;
    for (int r = 0; r < 64; ++r)
      if (sScore[r] > best) { best = sScore[r]; bi = r; }   // first max wins
    out[b] = (float)sTok[bi];
  }
}

extern "C" void kernel_launch(void* const* d_in, const int* in_sizes, int n_in,
                              void* d_out, int out_size, void* d_ws, size_t ws_size,
                              hipStream_t stream) {
  (void)in_sizes; (void)n_in; (void)out_size; (void)d_ws; (void)ws_size;
  const float* logits = (const float*)d_in[0];
  const int*   kArr   = (const int*)d_in[1];
  const float* pArr   = (const float*)d_in[2];
  float*       out    = (float*)d_out;
  RBLNTopKTopPSampler_kernel<<<BROWS, NTHREADS, 0, stream>>>(logits, kArr, pArr, out);
}